// DSS_Tripartite_12936441496030
// MI455X (gfx1250) — compile-verified
//
#include <hip/hip_runtime.h>
#include <hip/hip_bf16.h>

typedef float v2f __attribute__((ext_vector_type(2)));
typedef float v8f __attribute__((ext_vector_type(8)));

#define THREADS 256

// ---------------------------------------------------------------- zero
__global__ void zero_kernel(float* p, long long n) {
  long long i = (long long)blockIdx.x * blockDim.x + threadIdx.x;
  if (i < n) p[i] = 0.0f;
}

// ------------------------------------------------- rank-norm -> padded A (R x 4)
// pe_raw[:, :, 0]=rk_ui, 1=rk_bi, 2=1-rk_ui (valid only), col 3 = 0 (K pad)
__global__ void rank_pe_kernel(const int* __restrict__ bundle_items,
                               const float* __restrict__ ui_deg,
                               const float* __restrict__ bi_deg,
                               float* __restrict__ peA,
                               int NBU, int NT) {
  int b = blockIdx.x * blockDim.x + threadIdx.x;
  if (b >= NBU) return;
  const int MAXT = 32;
  float du[MAXT], db[MAXT];
  bool val[MAXT];
  int nvalid = 0;
  for (int t = 0; t < NT; ++t) {
    int it = bundle_items[b * NT + t];
    bool v = (it >= 0);
    val[t] = v;
    int ci = v ? it : 0;
    du[t] = v ? ui_deg[ci] : -1.0f;
    db[t] = v ? bi_deg[ci] : -1.0f;
    if (v) nvalid++;
  }
  float nv = (float)((nvalid > 2) ? nvalid : 2) - 1.0f;
  for (int t = 0; t < NT; ++t) {
    float o0 = 0.f, o1 = 0.f, o2 = 0.f;
    if (val[t]) {
      int ru = 0, rb = 0;
      float dut = du[t], dbt = db[t];
      for (int s = 0; s < NT; ++s) {  // stable argsort rank (desc deg, ties by idx)
        if (du[s] > dut || (du[s] == dut && s < t)) ru++;
        if (db[s] > dbt || (db[s] == dbt && s < t)) rb++;
      }
      float rku = fminf(fmaxf((float)ru / nv, 0.f), 1.f);
      float rkb = fminf(fmaxf((float)rb / nv, 0.f), 1.f);
      o0 = rku; o1 = rkb; o2 = 1.0f - rku;
    }
    float* p = &peA[(long long)(b * NT + t) * 4];
    p[0] = o0; p[1] = o1; p[2] = o2; p[3] = 0.f;
  }
}

// ------------------------------- WMMA: (R x 4) @ (4 x 64) -> scatter to global_pe
// One wave per 16-row tile; 4 column tiles of 16 cover D=64.
// A layout (16x4 f32): lane L: M=L%16, holds K = {2*(L/16), 2*(L/16)+1}
// B layout (4x16 f32): lane L: N=L%16, holds K rows {2*(L/16), 2*(L/16)+1}
// C/D layout: lane L: N=L%16; VGPR k: M = k + 8*(L/16)
__global__ void pe_wmma_scatter(const float* __restrict__ peA,
                                const float* __restrict__ Wpe,   // 3 x 64
                                const int* __restrict__ bundle_items, // flat R
                                float* __restrict__ gpe,          // NI x 64
                                int rowTiles, int R) {
  int wid  = (int)((blockIdx.x * blockDim.x + threadIdx.x) >> 5);
  int lane = threadIdx.x & 31;
  if (wid >= rowTiles) return;  // wave-uniform
  int half = lane >> 4;
  int m    = lane & 15;
  int k0   = 2 * half;
  int rowsBase = wid * 16;
  int rA = rowsBase + m;

  v2f a;
  a.x = peA[(long long)rA * 4 + k0];
  a.y = peA[(long long)rA * 4 + k0 + 1];

  int itemsK[8];
#pragma unroll
  for (int k = 0; k < 8; ++k) {
    int r2 = rowsBase + k + 8 * half;
    itemsK[k] = (r2 < R) ? bundle_items[r2] : -1;
  }

  for (int ct = 0; ct < 4; ++ct) {
    int col = ct * 16 + m;
    v2f b;
    b.x = Wpe[k0 * 64 + col];
    b.y = (k0 + 1 < 3) ? Wpe[(k0 + 1) * 64 + col] : 0.0f;
    v8f c = {};
#if defined(__gfx1250__) && __has_builtin(__builtin_amdgcn_wmma_f32_16x16x4_f32)
    c = __builtin_amdgcn_wmma_f32_16x16x4_f32(false, a, false, b, (short)0, c,
                                              false, false);
#else
#pragma unroll
    for (int k = 0; k < 8; ++k) {
      int r2 = rowsBase + k + 8 * half;
      if (r2 < R)
        c[k] = peA[(long long)r2 * 4 + 0] * Wpe[0 * 64 + col] +
               peA[(long long)r2 * 4 + 1] * Wpe[1 * 64 + col] +
               peA[(long long)r2 * 4 + 2] * Wpe[2 * 64 + col];
    }
#endif
#pragma unroll
    for (int k = 0; k < 8; ++k) {
      int it = itemsK[k];
      if (it >= 0) atomicAdd(&gpe[(long long)it * 64 + col], c[k]);
    }
  }
}

// ----------------------------------------- feat0 = concat(u, b, i + gpe/deg)
__global__ void build_feat(const float* __restrict__ uf, const float* __restrict__ bf,
                           const float* __restrict__ itf, const float* __restrict__ gpe,
                           const float* __restrict__ bi_deg,
                           float* __restrict__ feat, float* __restrict__ agg,
                           int NU, int NBU, int NI) {
  long long i = (long long)blockIdx.x * blockDim.x + threadIdx.x;
  long long total = (long long)(NU + NBU + NI) * 64;
  if (i >= total) return;
  int row = (int)(i >> 6), d = (int)(i & 63);
  float v;
  if (row < NU) {
    v = uf[i];
  } else if (row < NU + NBU) {
    v = bf[(long long)(row - NU) * 64 + d];
  } else {
    int ir = row - NU - NBU;
    float dv = fmaxf(bi_deg[ir], 1.0f);
    v = itf[(long long)ir * 64 + d] + gpe[(long long)ir * 64 + d] / dv;
  }
  feat[i] = v;
  agg[i] = v;
}

// ---------------------------------------------- SpMM: fout[row] += v*fin[col]
// 16 lanes per edge, float4 per lane (L2-resident gather+scatter on MI455X).
__global__ void spmm_kernel(const int* __restrict__ rows, const int* __restrict__ cols,
                            const float* __restrict__ vals,
                            const float* __restrict__ fin, float* __restrict__ fout,
                            int E) {
  long long tid = (long long)blockIdx.x * blockDim.x + threadIdx.x;
  int e = (int)(tid >> 4);
  if (e >= E) return;
  int j = ((int)tid & 15) * 4;
  int c = cols[e], r = rows[e];
  float v = vals[e];
  const float4 x = *(const float4*)(fin + (long long)c * 64 + j);
  float* o = fout + (long long)r * 64 + j;
  atomicAdd(o + 0, x.x * v);
  atomicAdd(o + 1, x.y * v);
  atomicAdd(o + 2, x.z * v);
  atomicAdd(o + 3, x.w * v);
}

// ---------------------------------------------- agg += l2norm(fin) (wave/row)
__global__ void l2norm_agg(const float* __restrict__ fin, float* __restrict__ agg,
                           int N) {
  int w = (int)((blockIdx.x * blockDim.x + threadIdx.x) >> 5);
  int lane = threadIdx.x & 31;
  if (w >= N) return;
  const float* p = fin + (long long)w * 64;
  float v0 = p[lane], v1 = p[lane + 32];
  float ss = v0 * v0 + v1 * v1;
  for (int m = 16; m >= 1; m >>= 1) ss += __shfl_xor(ss, m, 32);
  float inv = 1.0f / fmaxf(sqrtf(ss), 1e-12f);
  float* a = agg + (long long)w * 64;
  a[lane]      += v0 * inv;
  a[lane + 32] += v1 * inv;
}

// ---------------------------------------------- BPR: wave per batch row
__global__ void score_bpr(const float* __restrict__ agg, const int* __restrict__ users,
                          const int* __restrict__ bundles, float* __restrict__ out,
                          int NU, int BS, int NN1, float scoreScale, float meanScale) {
  int w = (int)((blockIdx.x * blockDim.x + threadIdx.x) >> 5);
  int lane = threadIdx.x & 31;
  if (w >= BS) return;
  int u = users[w];
  const float* au = agg + (long long)u * 64;
  float u0 = au[lane], u1 = au[lane + 32];
  float s[8];
  for (int n = 0; n < NN1 && n < 8; ++n) {
    int bi = bundles[w * NN1 + n];
    const float* ab = agg + (long long)(NU + bi) * 64;
    float d = u0 * ab[lane] + u1 * ab[lane + 32];
    for (int m = 16; m >= 1; m >>= 1) d += __shfl_xor(d, m, 32);
    s[n] = d * scoreScale;
  }
  if (lane == 0) {
    float pos = s[0], acc = 0.f;
    for (int n = 1; n < NN1 && n < 8; ++n) {
      float x = s[n] - pos;
      acc += fmaxf(x, 0.f) + log1pf(expf(-fabsf(x)));  // softplus
    }
    atomicAdd(out + 0, acc * meanScale);
  }
}

// ---------------------------------------------- reg = 0.5*(Σu0²+Σb0²+Σi0²)/BS
__global__ void reg_kernel(const float* __restrict__ uf, const float* __restrict__ bf,
                           const float* __restrict__ itf, const int* __restrict__ users,
                           const int* __restrict__ bundles, const int* __restrict__ bitems,
                           float* __restrict__ out, int NT, float scale,
                           long long nU, long long nB, long long nI) {
  long long i = (long long)blockIdx.x * blockDim.x + threadIdx.x;
  float v = 0.f;
  if (i < nU) {
    int s = (int)(i >> 6), d = (int)(i & 63);
    v = uf[(long long)users[s] * 64 + d];
  } else if (i < nU + nB) {
    long long r = i - nU;
    int s = (int)(r >> 6), d = (int)(r & 63);
    v = bf[(long long)bundles[s] * 64 + d];
  } else if (i < nU + nB + nI) {
    long long r = i - nU - nB;
    int d = (int)(r & 63);
    long long rt = r >> 6;
    int t = (int)(rt % NT);
    int s = (int)(rt / NT);
    int fb = bundles[s];
    int bi = bitems[(long long)fb * NT + t];
    v = (bi >= 0) ? itf[(long long)bi * 64 + d] : 0.0f;
  }
  float sq = v * v;
  __shared__ float red[THREADS];
  red[threadIdx.x] = sq;
  __syncthreads();
  for (int st = THREADS / 2; st >= 1; st >>= 1) {
    if (threadIdx.x < st) red[threadIdx.x] += red[threadIdx.x + st];
    __syncthreads();
  }
  if (threadIdx.x == 0) atomicAdd(out + 2, red[0] * scale);
}

static inline int gblocks(long long n, int t) { return (int)((n + t - 1) / t); }

extern "C" void kernel_launch(void* const* d_in, const int* in_sizes, int n_in,
                              void* d_out, int out_size, void* d_ws, size_t ws_size,
                              hipStream_t stream) {
  const float* uf       = (const float*)d_in[0];
  const float* bf       = (const float*)d_in[1];
  const float* itf      = (const float*)d_in[2];
  const float* Wpe      = (const float*)d_in[3];
  const float* ui_deg   = (const float*)d_in[4];
  const float* bi_deg   = (const float*)d_in[5];
  const float* tri_vals = (const float*)d_in[6];
  const int* bundle_items = (const int*)d_in[7];
  const int* tri_row    = (const int*)d_in[8];
  const int* tri_col    = (const int*)d_in[9];
  const int* users      = (const int*)d_in[10];
  const int* bundles    = (const int*)d_in[11];

  const int D = 64;
  const int NU  = in_sizes[0] / D;
  const int NBU = in_sizes[1] / D;
  const int NI  = in_sizes[2] / D;
  const int E   = in_sizes[6];
  const int NT  = in_sizes[7] / NBU;
  const int BS  = in_sizes[10];
  const int NN1 = in_sizes[11] / BS;
  const int N   = NU + NBU + NI;
  const int R   = NBU * NT;
  const int rowTiles = (R + 15) / 16;

  float* ws = (float*)d_ws;
  long long o = 0;
  float* peA  = ws + o; o += (long long)rowTiles * 16 * 4;
  float* gpe  = ws + o; o += (long long)NI * D;
  float* feat = ws + o; o += (long long)N * D;
  float* feat2= ws + o; o += (long long)N * D;
  float* agg  = ws + o;

  float* out = (float*)d_out;
  const long long ND = (long long)N * D;

  // init
  zero_kernel<<<1, 32, 0, stream>>>(out, 3);
  zero_kernel<<<gblocks((long long)NI * D, THREADS), THREADS, 0, stream>>>(gpe, (long long)NI * D);
  zero_kernel<<<gblocks((long long)rowTiles * 64, THREADS), THREADS, 0, stream>>>(peA, (long long)rowTiles * 64);

  // positional encoding: ranks -> WMMA GEMM -> scatter into global_pe
  rank_pe_kernel<<<gblocks(NBU, THREADS), THREADS, 0, stream>>>(bundle_items, ui_deg, bi_deg, peA, NBU, NT);
  pe_wmma_scatter<<<gblocks((long long)rowTiles * 32, 128), 128, 0, stream>>>(peA, Wpe, bundle_items, gpe, rowTiles, R);

  // feat0 and agg accumulator
  build_feat<<<gblocks(ND, THREADS), THREADS, 0, stream>>>(uf, bf, itf, gpe, bi_deg, feat, agg, NU, NBU, NI);

  // layer 1: feat2 = S*feat ; agg += l2norm(feat2)
  zero_kernel<<<gblocks(ND, THREADS), THREADS, 0, stream>>>(feat2, ND);
  spmm_kernel<<<gblocks((long long)E * 16, THREADS), THREADS, 0, stream>>>(tri_row, tri_col, tri_vals, feat, feat2, E);
  l2norm_agg<<<gblocks((long long)N * 32, THREADS), THREADS, 0, stream>>>(feat2, agg, N);

  // layer 2: feat = S*feat2 ; agg += l2norm(feat)
  zero_kernel<<<gblocks(ND, THREADS), THREADS, 0, stream>>>(feat, ND);
  spmm_kernel<<<gblocks((long long)E * 16, THREADS), THREADS, 0, stream>>>(tri_row, tri_col, tri_vals, feat2, feat, E);
  l2norm_agg<<<gblocks((long long)N * 32, THREADS), THREADS, 0, stream>>>(feat, agg, N);

  // scores & BPR (agg holds 3x the mean -> score scale 1/9)
  const float scoreScale = 1.0f / 9.0f;
  const float meanScale  = 1.0f / (float)(BS * (NN1 - 1));
  score_bpr<<<gblocks((long long)BS * 32, THREADS), THREADS, 0, stream>>>(agg, users, bundles, out, NU, BS, NN1, scoreScale, meanScale);

  // regularizer (out[1] stays 0 from the zero kernel)
  const long long nUe = (long long)BS * 64;
  const long long nBe = (long long)BS * NN1 * 64;
  const long long nIe = (long long)BS * NN1 * NT * 64;
  reg_kernel<<<gblocks(nUe + nBe + nIe, THREADS), THREADS, 0, stream>>>(
      uf, bf, itf, users, bundles, bundle_items, out, NT, 0.5f / (float)BS, nUe, nBe, nIe);
}